// TemporalCRF_64699387347417
// MI455X (gfx1250) — compile-verified
//
#include <hip/hip_runtime.h>
#include <stdint.h>

// Problem constants (from reference): B=256, T=16384, C=8
#define Bsz    256
#define Tlen   16384
#define NCHUNK 64      // time chunks per batch
#define CHLEN  256     // steps per chunk (NCHUNK*CHLEN == Tlen)
#define KB     64      // steps staged per async double-buffer block
#define NBLK   (CHLEN / KB)

typedef __attribute__((ext_vector_type(2))) float v2f;
typedef __attribute__((ext_vector_type(8))) float v8f;

// ds_swizzle group-of-32: offset = {xor[14:10], or[9:5], and[4:0]}
__device__ __forceinline__ float swz_bcast16(float x) {        // src = lane & 15
  return __int_as_float(__builtin_amdgcn_ds_swizzle(__float_as_int(x), 0x000F));
}
__device__ __forceinline__ float swz_swap16(float x) {         // src = lane ^ 16
  return __int_as_float(__builtin_amdgcn_ds_swizzle(__float_as_int(x), 0x401F));
}

// ---------------------------------------------------------------------------
// Kernel 1: per-(batch-pair, chunk) forward-matrix products in scaled-exp
// domain, driven by v_wmma_f32_16x16x4_f32; emissions streamed via
// global_load_async_to_lds (double buffered). Also accumulates the fused
// path-score partials so emissions are read from HBM exactly once.
// Wave layout: 16 matrix columns = 2 batches x 8 basis vectors.
// ---------------------------------------------------------------------------
__global__ __launch_bounds__(32) void crf_chunk_kernel(
    const float* __restrict__ em, const int* __restrict__ tags,
    const unsigned char* __restrict__ mask, const float* __restrict__ trans,
    float* __restrict__ Mout, float* __restrict__ LamOut,
    float* __restrict__ scoreP, float* __restrict__ mcntP)
{
  __shared__ float          emT[2][2][KB][8];   // [buf][b_local][t][state]
  __shared__ unsigned char  mkT[2][2][KB];      // [buf][b_local][t]
  __shared__ float          trS[64];

  const int  lane  = threadIdx.x;
  const bool lt16  = lane < 16;
  const int  b0    = blockIdx.x * 2;
  const int  c     = blockIdx.y;
  const long tbase = (long)c * CHLEN;

  trS[lane]      = trans[lane];
  trS[lane + 32] = trans[lane + 32];
  __syncthreads();

  // Constant A operands: A = E^T padded to 16 rows (rows >= 8 are zero).
  // A layout (16x4 f32): lane<16 holds row=lane, k={0,1}; lane>=16 row=lane-16, k={2,3}.
  v2f a1, a2;
  {
    const int j    = lane & 15;
    const int krow = lt16 ? 0 : 2;
    float e0 = 0.f, e1 = 0.f, e2 = 0.f, e3 = 0.f;
    if (j < 8) {
      e0 = __expf(trans[(krow + 0) * 8 + j]);   // E[k][j] = exp(trans[k][j])
      e1 = __expf(trans[(krow + 1) * 8 + j]);
      e2 = __expf(trans[(krow + 4) * 8 + j]);
      e3 = __expf(trans[(krow + 5) * 8 + j]);
    }
    a1 = (v2f){e0, e1};
    a2 = (v2f){e2, e3};
  }

  // State X (8x16) kept in B layout (4x16 per WMMA):
  //   b1x: lane<16 -> X[0][col], lane>=16 -> X[2][col]   (col = lane & 15)
  //   b1y: X[1]/X[3];  b2x: X[4]/X[6];  b2y: X[5]/X[7]
  const int col = lane & 15;
  const int rr  = col & 7;            // basis index of this column
  const int blc = col >> 3;           // batch (0/1) of this column
  float b1x = (lt16 ? (rr == 0) : (rr == 2)) ? 1.f : 0.f;   // X = identity
  float b1y = (lt16 ? (rr == 1) : (rr == 3)) ? 1.f : 0.f;
  float b2x = (lt16 ? (rr == 4) : (rr == 6)) ? 1.f : 0.f;
  float b2y = (lt16 ? (rr == 5) : (rr == 7)) ? 1.f : 0.f;
  float lam2 = 0.f;                   // accumulated log2 renorm scale (per column)

  float scAcc = 0.f, mcAcc = 0.f;     // fused path-score / mask-count partials

  auto stage = [&](int buf, int blk) {
    const long t0 = tbase + (long)blk * KB;
    // emissions: contiguous 2048B per batch -> 4 x b128 wave issues per batch
    for (int bl = 0; bl < 2; ++bl) {
      const float* gp = em + ((long)(b0 + bl) * Tlen + t0) * 8;
      unsigned lbase = (unsigned)(uintptr_t)&emT[buf][bl][0][0];
      unsigned long long gbase = (unsigned long long)(uintptr_t)gp;
      for (int q = 0; q < 4; ++q) {
        unsigned           la = lbase + q * 512 + lane * 16;
        unsigned long long ga = gbase + q * 512 + lane * 16;
        asm volatile("global_load_async_to_lds_b128 %0, %1, off"
                     :: "v"(la), "v"(ga) : "memory");
      }
    }
    // mask bytes: 64B per batch -> one b32 wave issue covers both batches
    {
      const int bl = lane >> 4;
      unsigned long long ga = (unsigned long long)(uintptr_t)
          (mask + (long)(b0 + bl) * Tlen + t0 + (long)(lane & 15) * 4);
      unsigned la = (unsigned)(uintptr_t)&mkT[buf][bl][(lane & 15) * 4];
      asm volatile("global_load_async_to_lds_b32 %0, %1, off"
                   :: "v"(la), "v"(ga) : "memory");
    }
  };

  auto process = [&](int buf, int blk) {
    const long t0 = tbase + (long)blk * KB;
    const int  istart = (t0 == 0) ? 1 : 0;   // step t==0 does not exist

    // ---- fused path-score partial (lane mapping: batch = lane>>4, 4 t's/lane)
    // Must run BEFORE the in-place exp() transform below (needs raw emissions).
    {
      const int  bl   = lane >> 4;
      const long brow = (long)(b0 + bl) * Tlen;
      for (int k = 0; k < 4; ++k) {
        const int  tl = (lane & 15) + k * 16;
        const long t  = t0 + tl;
        if (t >= 1) {
          if (mkT[buf][bl][tl]) {
            const int tg = tags[brow + t];
            const int tp = tags[brow + t - 1];
            scAcc += emT[buf][bl][tl][tg] + trS[tp * 8 + tg];
            mcAcc += 1.f;
          }
        }
      }
    }

    // ---- transform tile in place: emT <- exp(emT)  (32 exps/lane, off-chain)
    {
      float* base = &emT[buf][0][0][0];        // 1024 floats per buffer
      #pragma unroll
      for (int q = 0; q < 8; ++q) {
        float4* p = (float4*)(base + q * 128 + lane * 4);
        float4 v = *p;
        v.x = __expf(v.x); v.y = __expf(v.y); v.z = __expf(v.z); v.w = __expf(v.w);
        *p = v;
      }
    }

    // ---- WMMA recurrence: X <- diag(w_t) * E^T * X, 8 steps per renorm group
    for (int g = 0; g < KB / 8; ++g) {
      const unsigned long long mk8 =
          *(const unsigned long long*)&mkT[buf][blc][g * 8];
      #pragma unroll
      for (int ii = 0; ii < 8; ++ii) {
        const int i = g * 8 + ii;
        const float4 wa = *(const float4*)&emT[buf][blc][i][0];
        const float4 wb = *(const float4*)&emT[buf][blc][i][4];
        const bool mv = (i >= istart) && (((mk8 >> (ii * 8)) & 0xFFull) != 0);

        v8f acc = {};
        acc = __builtin_amdgcn_wmma_f32_16x16x4_f32(false, a1, false, (v2f){b1x, b1y},
                                                    (short)0, acc, false, false);
        acc = __builtin_amdgcn_wmma_f32_16x16x4_f32(false, a2, false, (v2f){b2x, b2y},
                                                    (short)0, acc, false, false);
        // Scale rows by w (valid data lives in lanes 0..15 of D layout)
        const float s0 = acc[0] * wa.x, s1 = acc[1] * wa.y;
        const float s2 = acc[2] * wa.z, s3 = acc[3] * wa.w;
        const float s4 = acc[4] * wb.x, s5 = acc[5] * wb.y;
        const float s6 = acc[6] * wb.z, s7 = acc[7] * wb.w;
        // Repack D -> B layout. NOTE: swizzles are hoisted and unconditional so
        // they execute with full EXEC (convergent op); select afterwards.
        const float u2 = swz_bcast16(s2);
        const float u3 = swz_bcast16(s3);
        const float u6 = swz_bcast16(s6);
        const float u7 = swz_bcast16(s7);
        const float n1x = lt16 ? s0 : u2;
        const float n1y = lt16 ? s1 : u3;
        const float n2x = lt16 ? s4 : u6;
        const float n2y = lt16 ? s5 : u7;
        // Masked step == identity: keep previous state
        b1x = mv ? n1x : b1x;
        b1y = mv ? n1y : b1y;
        b2x = mv ? n2x : b2x;
        b2y = mv ? n2y : b2y;
      }
      // exact power-of-2 renorm per column (both lane halves agree)
      {
        const float m4 = fmaxf(fmaxf(b1x, b1y), fmaxf(b2x, b2y));
        const float mo = swz_swap16(m4);
        const float m  = fmaxf(m4, mo);
        const int   k  = ((__float_as_int(m) >> 23) & 0xFF) - 127;
        const float sc = __int_as_float((127 - k) << 23);  // 2^-k, exact
        b1x *= sc; b1y *= sc; b2x *= sc; b2y *= sc;
        lam2 += (float)k;
      }
    }
  };

  stage(0, 0);
  for (int blk = 0; blk < NBLK; ++blk) {
    if (blk + 1 < NBLK) {
      stage((blk + 1) & 1, blk + 1);
      asm volatile("s_wait_asynccnt 9" ::: "memory");   // current buffer's 9 ops done
    } else {
      asm volatile("s_wait_asynccnt 0" ::: "memory");
    }
    process(blk & 1, blk);
  }

  // ---- write chunk matrix M[b][c][j][r] (scaled) and per-column log2 scale
  {
    const int bl = col >> 3, r = col & 7;
    const long base = ((long)(b0 + bl) * NCHUNK + c) * 64;
    const int j0 = lt16 ? 0 : 2;
    const int j2 = lt16 ? 4 : 6;
    Mout[base + (j0 + 0) * 8 + r] = b1x;
    Mout[base + (j0 + 1) * 8 + r] = b1y;
    Mout[base + (j2 + 0) * 8 + r] = b2x;
    Mout[base + (j2 + 1) * 8 + r] = b2y;
    if (lt16) LamOut[((long)(b0 + bl) * NCHUNK + c) * 8 + r] = lam2;
  }
  // ---- reduce score partials across the 16 lanes of each batch group
  for (int d = 1; d < 16; d <<= 1) {
    scAcc += __shfl_xor(scAcc, d, 32);
    mcAcc += __shfl_xor(mcAcc, d, 32);
  }
  if ((lane & 15) == 0) {
    const long idx = (long)(b0 + (lane >> 4)) * NCHUNK + c;
    scoreP[idx] = scAcc;
    mcntP[idx]  = mcAcc;
  }
}

// ---------------------------------------------------------------------------
// Kernel 2: per-batch combine. Folds the 64 chunk matrices in log domain,
// finishes the path score, emits out[b] = logZ - score. One wave per batch.
// ---------------------------------------------------------------------------
__global__ __launch_bounds__(32) void crf_combine_kernel(
    const float* __restrict__ em, const int* __restrict__ tags,
    const unsigned char* __restrict__ mask,
    const float* __restrict__ startT, const float* __restrict__ endT,
    const float* __restrict__ Mout, const float* __restrict__ Lam,
    const float* __restrict__ scoreP, const float* __restrict__ mcntP,
    float* __restrict__ out)
{
  const int b = blockIdx.x;
  const int lane = threadIdx.x;
  const float LN2 = 0.6931471805599453f;

  // Sum fused score / mask-count partials
  float sc = 0.f, mc = 0.f;
  for (int i = lane; i < NCHUNK; i += 32) {
    sc += scoreP[(long)b * NCHUNK + i];
    mc += mcntP[(long)b * NCHUNK + i];
  }
  for (int d = 16; d >= 1; d >>= 1) {
    sc += __shfl_xor(sc, d, 32);
    mc += __shfl_xor(mc, d, 32);
  }
  const int lastv = (int)(mc + (mask[(long)b * Tlen] ? 1.f : 0.f)) - 1;
  const int tag0  = tags[(long)b * Tlen];
  const int tagL  = tags[(long)b * Tlen + lastv];
  const float score = sc + startT[tag0] + em[(long)b * Tlen * 8 + tag0] + endT[tagL];

  // alpha propagation (log domain), lanes 0..7 hold alpha[j]
  const int j = (lane < 8) ? lane : 0;
  float alpha = startT[j] + em[(long)b * Tlen * 8 + j];
  for (int c = 0; c < NCHUNK; ++c) {
    const float* Mr = Mout + (((long)b * NCHUNK + c) * 8 + j) * 8;
    const float* Lr = Lam  +  ((long)b * NCHUNK + c) * 8;
    float tm[8];
    float mx = -INFINITY;
    #pragma unroll
    for (int r = 0; r < 8; ++r) {
      const float ar = __shfl(alpha, r, 32);
      const float t  = ar + Lr[r] * LN2 + __logf(Mr[r]);  // log(0) -> -inf ok
      tm[r] = t;
      mx = fmaxf(mx, t);
    }
    float s = 0.f;
    #pragma unroll
    for (int r = 0; r < 8; ++r) s += __expf(tm[r] - mx);
    const float an = mx + __logf(s);
    if (lane < 8) alpha = an;
  }

  // logZ = logsumexp_j(alpha + endT)
  float v = (lane < 8) ? (alpha + endT[j]) : -INFINITY;
  float mx = v;
  for (int d = 4; d >= 1; d >>= 1) mx = fmaxf(mx, __shfl_xor(mx, d, 32));
  float e = (lane < 8) ? __expf(v - mx) : 0.f;
  for (int d = 4; d >= 1; d >>= 1) e += __shfl_xor(e, d, 32);
  if (lane == 0) out[b] = (mx + __logf(e)) - score;   // -(score - logZ)
}

// ---------------------------------------------------------------------------
extern "C" void kernel_launch(void* const* d_in, const int* in_sizes, int n_in,
                              void* d_out, int out_size, void* d_ws, size_t ws_size,
                              hipStream_t stream) {
  (void)in_sizes; (void)n_in; (void)out_size; (void)ws_size;
  const float*         em     = (const float*)d_in[0];
  const int*           tags   = (const int*)d_in[1];
  const unsigned char* mask   = (const unsigned char*)d_in[2];
  const float*         trans  = (const float*)d_in[3];
  const float*         startT = (const float*)d_in[4];
  const float*         endT   = (const float*)d_in[5];
  float*               out    = (float*)d_out;

  // Workspace partition (floats): M[B][NC][8][8], Lam[B][NC][8], score/mcnt[B][NC]
  float* Mout   = (float*)d_ws;
  float* Lam    = Mout   + (size_t)Bsz * NCHUNK * 64;
  float* scoreP = Lam    + (size_t)Bsz * NCHUNK * 8;
  float* mcntP  = scoreP + (size_t)Bsz * NCHUNK;

  dim3 g1(Bsz / 2, NCHUNK);
  hipLaunchKernelGGL(crf_chunk_kernel, g1, dim3(32), 0, stream,
                     em, tags, mask, trans, Mout, Lam, scoreP, mcntP);
  hipLaunchKernelGGL(crf_combine_kernel, dim3(Bsz), dim3(32), 0, stream,
                     em, tags, mask, startT, endT, Mout, Lam, scoreP, mcntP, out);
}